// PriorKnowledgeLayer_51007031608223
// MI455X (gfx1250) — compile-verified
//
#include <hip/hip_runtime.h>

typedef __attribute__((ext_vector_type(2))) float v2f;
typedef __attribute__((ext_vector_type(4))) float v4f;
typedef __attribute__((ext_vector_type(8))) float v8f;

#define BB 128
#define SS 4096
#define CC 64
#define TB 128          // sequence pairs per block
#define PPITCH 65       // LDS pitch for p tile (bank-conflict-free WMMA A reads)
#define BPR 160         // floats per k2-row of paired-B buffer (160 % 64 == 32 -> halves disjoint)

__global__ __launch_bounds__(256) void pkl_kernel(const float* __restrict__ logits,
                                                  const float* __restrict__ Mg,
                                                  float* __restrict__ out) {
    __shared__ float Pt[(TB + 1) * PPITCH];   // log-softmax rows i0 .. i0+TB
    __shared__ float BP[(CC / 2) * BPR];      // BP[k2][n] pair = {M[n][2k2], M[n][2k2+1]}
    __shared__ float OutBuf[TB];

    const int tid  = threadIdx.x;
    const int lane = tid & 31;
    const int wave = tid >> 5;
    const int l16  = lane & 15;
    const int half = lane >> 4;

    const int chunk = blockIdx.x;
    const int b     = blockIdx.y;
    const int i0    = chunk * TB;
    const int npair = min(TB, (SS - 1) - i0);   // valid output rows in this block
    const int nrows = npair + 1;                // softmax rows needed (incl. i+1 boundary)

    const float* gbase = logits + ((size_t)b * SS + i0) * CC;

    // ---- Stage logits tile into LDS: b128 global loads, padded-pitch scatter ----
    for (int idx4 = tid; idx4 < (TB + 1) * CC / 4; idx4 += 256) {
        int r  = idx4 >> 4;            // 16 float4 per 64-wide row
        int c  = (idx4 & 15) * 4;
        v4f v = {0.0f, 0.0f, 0.0f, 0.0f};
        if (r < nrows) v = *(const v4f*)(gbase + idx4 * 4);
        float* p = &Pt[r * PPITCH + c];
        p[0] = v.x; p[1] = v.y; p[2] = v.z; p[3] = v.w;
    }
    // ---- Stage M as paired-k B operand: BP[k2][n] = {M[n][2k2], M[n][2k2+1]} ----
    for (int idx = tid; idx < CC * CC; idx += 256) {
        int n = idx >> 6, k = idx & 63;
        BP[(k >> 1) * BPR + n * 2 + (k & 1)] = Mg[idx];
    }
    __syncthreads();

    // ---- In-place log_softmax: one wave per row, 2 elements per lane ----
    for (int r = wave; r < nrows; r += 8) {
        float x0 = Pt[r * PPITCH + lane];
        float x1 = Pt[r * PPITCH + lane + 32];
        float m = fmaxf(x0, x1);
        for (int s = 16; s > 0; s >>= 1) m = fmaxf(m, __shfl_xor(m, s, 32));
        float e = __expf(x0 - m) + __expf(x1 - m);
        for (int s = 16; s > 0; s >>= 1) e += __shfl_xor(e, s, 32);
        float ls = m + __logf(e);
        Pt[r * PPITCH + lane]      = x0 - ls;
        Pt[r * PPITCH + lane + 32] = x1 - ls;
    }
    __syncthreads();

    // ---- WMMA: T[r][c] = sum_d p[i0+r+1][d] * M[c][d], 16x64 tile per wave ----
    const int r0   = wave * 16;
    const int rowA = r0 + l16 + 1;     // A rows come from the "next" position
    v8f acc0 = {}, acc1 = {}, acc2 = {}, acc3 = {};

#pragma unroll
    for (int k0 = 0; k0 < CC; k0 += 4) {
        const int ka  = k0 + 2 * half;        // A lane K assignment (lanes16-31: K+2,K+3)
        const int k2a = (k0 >> 1) + half;     // paired-B row
        v2f a;
        a.x = Pt[rowA * PPITCH + ka];
        a.y = Pt[rowA * PPITCH + ka + 1];

        const float* brow = &BP[k2a * BPR];
        v2f b0 = *(const v2f*)(brow + (l16 +  0) * 2);
        v2f b1 = *(const v2f*)(brow + (l16 + 16) * 2);
        v2f b2 = *(const v2f*)(brow + (l16 + 32) * 2);
        v2f b3 = *(const v2f*)(brow + (l16 + 48) * 2);

        acc0 = __builtin_amdgcn_wmma_f32_16x16x4_f32(false, a, false, b0, (short)0, acc0, false, false);
        acc1 = __builtin_amdgcn_wmma_f32_16x16x4_f32(false, a, false, b1, (short)0, acc1, false, false);
        acc2 = __builtin_amdgcn_wmma_f32_16x16x4_f32(false, a, false, b2, (short)0, acc2, false, false);
        acc3 = __builtin_amdgcn_wmma_f32_16x16x4_f32(false, a, false, b3, (short)0, acc3, false, false);
    }

    // ---- loss[row] = sum_c p_cur[row][c] * T[row][c]; 16-lane reduction ----
#pragma unroll
    for (int v = 0; v < 8; ++v) {
        const int row = r0 + v + 8 * half;   // C/D layout: VGPR v -> rows v / v+8
        float sv = acc0[v] * Pt[row * PPITCH + l16 +  0]
                 + acc1[v] * Pt[row * PPITCH + l16 + 16]
                 + acc2[v] * Pt[row * PPITCH + l16 + 32]
                 + acc3[v] * Pt[row * PPITCH + l16 + 48];
        sv += __shfl_xor(sv, 1, 32);
        sv += __shfl_xor(sv, 2, 32);
        sv += __shfl_xor(sv, 4, 32);
        sv += __shfl_xor(sv, 8, 32);
        if (l16 == 0) OutBuf[row] = sv;
    }
    __syncthreads();

    // ---- Coalesced output store ----
    if (tid < npair) {
        out[(size_t)b * (SS - 1) + i0 + tid] = OutBuf[tid];
    }
}

extern "C" void kernel_launch(void* const* d_in, const int* in_sizes, int n_in,
                              void* d_out, int out_size, void* d_ws, size_t ws_size,
                              hipStream_t stream) {
    const float* logits = (const float*)d_in[0];
    const float* Mg     = (const float*)d_in[1];
    float* out          = (float*)d_out;

    dim3 grid((SS - 1 + TB - 1) / TB, BB);   // 32 x 128 blocks
    pkl_kernel<<<grid, 256, 0, stream>>>(logits, Mg, out);
}